// GNN_19619410608395
// MI455X (gfx1250) — compile-verified
//
#include <hip/hip_runtime.h>

typedef float v2f __attribute__((ext_vector_type(2)));
typedef float v8f __attribute__((ext_vector_type(8)));

#define LRELU_SLOPE 0.01f

// ---------------- degree / normalization ----------------

__global__ __launch_bounds__(256) void gcn_deg_init(float* __restrict__ deg, int n) {
  int i = blockIdx.x * 256 + threadIdx.x;
  if (i < n) deg[i] = 1.0f;  // self loop contributes 1 to every node's degree
}

__global__ __launch_bounds__(256) void gcn_deg_edges(float* __restrict__ deg,
                                                     const int* __restrict__ dst, int e) {
  int i = blockIdx.x * 256 + threadIdx.x;
  if (i < e)
    __hip_atomic_fetch_add(&deg[dst[i]], 1.0f, __ATOMIC_RELAXED, __HIP_MEMORY_SCOPE_AGENT);
}

__global__ __launch_bounds__(256) void gcn_deg_rsqrt(float* __restrict__ deg, int n) {
  int i = blockIdx.x * 256 + threadIdx.x;
  if (i < n) {
    float d = deg[i];
    deg[i] = d > 0.0f ? rsqrtf(d) : 0.0f;
  }
}

// ---------------- GEMM: H[n x cout] = X[n x 256] @ W[cout x 256]^T ----------------
// fp32 WMMA 16x16x4. One wave -> 16 rows x 64 cols (4 accumulator tiles share one A frag).
// W tile (64 cols x 256 K) staged in 64KB LDS with a rotate swizzle to avoid bank conflicts.

__global__ __launch_bounds__(256) void gcn_gemm(const float* __restrict__ X,
                                                const float* __restrict__ W,
                                                float* __restrict__ H,
                                                int nrows, int cout) {
  __shared__ float ldsW[64 * 256];  // exactly 64 KB

  const int tid  = threadIdx.x;
  const int col0 = blockIdx.y * 64;

  // Stage W[col0 .. col0+63][0..255] into LDS, swizzled: element (c,k) at c*256 + ((k+4c)&255).
  // Keeps b64 pairs adjacent (offsets even) and spreads the 16 half-wave lanes over banks.
  for (int i = tid; i < 64 * 256; i += 256) {
    int c = i >> 8;
    int k = i & 255;
    ldsW[(c << 8) | ((k + 4 * c) & 255)] = W[(size_t)(col0 + c) * 256 + k];
  }
  __syncthreads();

  const int wave = tid >> 5;
  const int lane = tid & 31;
  const int half = lane >> 4;   // 0: K={0,1}, 1: K={2,3} within each 4-step
  const int l    = lane & 15;
  const int row0 = (blockIdx.x * 8 + wave) * 16;
  if (row0 >= nrows) return;    // wave-uniform: EXEC stays all-ones for WMMA

  const float* xrow = X + (size_t)(row0 + l) * 256;

  const int c0 = l, c1 = 16 + l, c2 = 32 + l, c3 = 48 + l;
  const int r0 = (4 * c0) & 255, r1 = (4 * c1) & 255,
            r2 = (4 * c2) & 255, r3 = (4 * c3) & 255;
  const float* w0 = ldsW + (c0 << 8);
  const float* w1 = ldsW + (c1 << 8);
  const float* w2 = ldsW + (c2 << 8);
  const float* w3 = ldsW + (c3 << 8);

  v8f acc0 = {}, acc1 = {}, acc2 = {}, acc3 = {};

  for (int k = 0; k < 256; k += 4) {
    const int kk = k + half * 2;
    // A 16x4 f32 frag: lanes 0-15 hold (row, K=kk..kk+1), lanes 16-31 hold K=kk+2..kk+3
    v2f a = *(const v2f*)(xrow + kk);
    // B 4x16 f32 frags: lane's column c, same K split as A (B row striped across lanes)
    v2f b0 = *(const v2f*)(w0 + ((kk + r0) & 255));
    v2f b1 = *(const v2f*)(w1 + ((kk + r1) & 255));
    v2f b2 = *(const v2f*)(w2 + ((kk + r2) & 255));
    v2f b3 = *(const v2f*)(w3 + ((kk + r3) & 255));
    acc0 = __builtin_amdgcn_wmma_f32_16x16x4_f32(false, a, false, b0, (short)0, acc0, false, false);
    acc1 = __builtin_amdgcn_wmma_f32_16x16x4_f32(false, a, false, b1, (short)0, acc1, false, false);
    acc2 = __builtin_amdgcn_wmma_f32_16x16x4_f32(false, a, false, b2, (short)0, acc2, false, false);
    acc3 = __builtin_amdgcn_wmma_f32_16x16x4_f32(false, a, false, b3, (short)0, acc3, false, false);
  }

  // C/D layout: VGPR v -> row (row0 + v + 8*half), column = col0 + tile*16 + l
#pragma unroll
  for (int v = 0; v < 8; ++v) {
    float* hp = H + (size_t)(row0 + v + half * 8) * cout + col0 + l;
    hp[0]  = acc0[v];
    hp[16] = acc1[v];
    hp[32] = acc2[v];
    hp[48] = acc3[v];
  }
}

// ---------------- aggregation ----------------

// out[i*C + c] = dinv[i]^2 * h[i*C + c]   (self-loop term; fully initializes the output buffer)
__global__ __launch_bounds__(256) void gcn_self(const float* __restrict__ h,
                                                const float* __restrict__ dinv,
                                                float* __restrict__ out,
                                                int total, int cshift) {
  int i = blockIdx.x * 256 + threadIdx.x;
  if (i >= total) return;
  float di = dinv[i >> cshift];
  out[i] = di * di * h[i];
}

// one wave per edge: out[dst] += dinv[src]*dinv[dst] * h[src]
__global__ __launch_bounds__(256) void gcn_scatter(const float* __restrict__ h,
                                                   const int* __restrict__ src,
                                                   const int* __restrict__ dst,
                                                   const float* __restrict__ dinv,
                                                   float* __restrict__ out,
                                                   int e, int C) {
  int w    = (blockIdx.x * 256 + threadIdx.x) >> 5;
  int lane = threadIdx.x & 31;
  if (w >= e) return;
  int s = src[w], d = dst[w];
  float nrm = dinv[s] * dinv[d];
  const float* hs = h + (size_t)s * C;
  float* od = out + (size_t)d * C;
  for (int c = lane; c < C; c += 32)
    __hip_atomic_fetch_add(od + c, nrm * hs[c], __ATOMIC_RELAXED, __HIP_MEMORY_SCOPE_AGENT);
}

// out = (out + b)  [optionally leaky-relu], in place
__global__ __launch_bounds__(256) void gcn_bias_act(float* __restrict__ out,
                                                    const float* __restrict__ b,
                                                    int total, int cmask, int act) {
  int i = blockIdx.x * 256 + threadIdx.x;
  if (i >= total) return;
  float v = out[i] + b[i & cmask];
  if (act) v = v > 0.0f ? v : LRELU_SLOPE * v;
  out[i] = v;
}

// ---------------- launch ----------------

extern "C" void kernel_launch(void* const* d_in, const int* in_sizes, int n_in,
                              void* d_out, int out_size, void* d_ws, size_t ws_size,
                              hipStream_t stream) {
  const float* x  = (const float*)d_in[0];
  const int*   ei = (const int*)d_in[1];
  const float* W1 = (const float*)d_in[2];
  const float* b1 = (const float*)d_in[3];
  const float* W2 = (const float*)d_in[4];
  const float* b2 = (const float*)d_in[5];
  const float* W3 = (const float*)d_in[6];
  const float* b3 = (const float*)d_in[7];

  const int N = in_sizes[0] / 256;  // 50000
  const int E = in_sizes[1] / 2;    // 800000
  const int* src = ei;
  const int* dst = ei + E;

  // workspace: dinv[N] | h[N*256] (pre-aggregation GEMM output, reused per layer)
  float* dinv = (float*)d_ws;
  size_t off = (((size_t)N * sizeof(float)) + 255) & ~(size_t)255;
  float* h = (float*)((char*)d_ws + off);

  float* h1     = (float*)d_out;             // [N,256]
  float* h2     = h1 + (size_t)N * 256;      // [N,256]
  float* logits = h2 + (size_t)N * 256;      // [N,64]

  const int nblk   = (N + 255) / 256;
  const int eblk   = (E + 255) / 256;
  const int tot256 = N * 256, tot64 = N * 64;
  const int nblk256 = (tot256 + 255) / 256, nblk64 = (tot64 + 255) / 256;
  const int sblk = (E * 32 + 255) / 256;          // one wave per edge
  const dim3 gemm_grid_wide((N + 127) / 128, 4);  // 8 waves/blk * 16 rows, 4 x 64-col tiles
  const dim3 gemm_grid_narrow((N + 127) / 128, 1);

  // normalization coefficients
  gcn_deg_init <<<nblk, 256, 0, stream>>>(dinv, N);
  gcn_deg_edges<<<eblk, 256, 0, stream>>>(dinv, dst, E);
  gcn_deg_rsqrt<<<nblk, 256, 0, stream>>>(dinv, N);

  // ---- layer 1: x -> h1 ----
  gcn_gemm    <<<gemm_grid_wide, 256, 0, stream>>>(x, W1, h, N, 256);
  gcn_self    <<<nblk256, 256, 0, stream>>>(h, dinv, h1, tot256, 8);
  gcn_scatter <<<sblk, 256, 0, stream>>>(h, src, dst, dinv, h1, E, 256);
  gcn_bias_act<<<nblk256, 256, 0, stream>>>(h1, b1, tot256, 255, 1);

  // ---- layer 2: h1 -> h2 ----
  gcn_gemm    <<<gemm_grid_wide, 256, 0, stream>>>(h1, W2, h, N, 256);
  gcn_self    <<<nblk256, 256, 0, stream>>>(h, dinv, h2, tot256, 8);
  gcn_scatter <<<sblk, 256, 0, stream>>>(h, src, dst, dinv, h2, E, 256);
  gcn_bias_act<<<nblk256, 256, 0, stream>>>(h2, b2, tot256, 255, 1);

  // ---- layer 3: h2 -> logits (no activation) ----
  gcn_gemm    <<<gemm_grid_narrow, 256, 0, stream>>>(h2, W3, h, N, 64);
  gcn_self    <<<nblk64, 256, 0, stream>>>(h, dinv, logits, tot64, 6);
  gcn_scatter <<<sblk, 256, 0, stream>>>(h, src, dst, dinv, logits, E, 64);
  gcn_bias_act<<<nblk64, 256, 0, stream>>>(logits, b3, tot64, 63, 0);
}